// FaceTokenizerANN_23244363006486
// MI455X (gfx1250) — compile-verified
//
#include <hip/hip_runtime.h>
#include <hip/hip_bf16.h>

typedef __attribute__((ext_vector_type(2))) float v2f;
typedef __attribute__((ext_vector_type(8))) float v8f;

#define NBATCH 64
#define TLEN   750
#define DIN    58
#define DOUT   58
#define DMODEL 256
#define NCODE  512
#define BT     (NBATCH * TLEN)        // 48000
#define MTILES (BT / 16)              // 3000
#define ENC_PER_BATCH (TLEN * DMODEL) // 192000
#define DEC_PER_BATCH (TLEN * DOUT)   // 43500
#define ENC_CHUNKS 25
#define ENC_PER_CHUNK (ENC_PER_BATCH / ENC_CHUNKS) // 7680
#define DEC_CHUNKS 15
#define DEC_PER_CHUNK (DEC_PER_BATCH / DEC_CHUNKS) // 2900
#define YTOTAL (BT * DOUT)            // 2784000

__device__ __forceinline__ v8f wmma_f32(v2f a, v2f b, v8f c) {
    return __builtin_amdgcn_wmma_f32_16x16x4_f32(false, a, false, b, (short)0, c, false, false);
}

// ---------------------------------------------------------------------------
// Generic wave-level GEMM: each wave computes a 16x64 strip of out = relu(A@B + bias).
// A is MxK row-major (optionally row-gathered through gidx), B is KxN row-major.
// KPAD is K rounded up to a multiple of 4; k >= KMAX reads are zero-filled.
// ---------------------------------------------------------------------------
template <int KMAX, int KPAD, bool GATHER>
__global__ __launch_bounds__(256) void gemm_relu_kernel(
    const float* __restrict__ A, const int* __restrict__ gidx,
    const float* __restrict__ Bw, const float* __restrict__ bias,
    float* __restrict__ out,
    int lda, int ldb, int ldo, int ncols, int ncolsB, int ngroups)
{
    int wid  = blockIdx.x * (blockDim.x >> 5) + (threadIdx.x >> 5);
    int mt   = wid / ngroups;
    int ng   = wid - mt * ngroups;
    int r0   = mt * 16;
    int n0   = ng * 64;
    int lane = threadIdx.x & 31;
    int half = lane >> 4;
    int lm   = lane & 15;

    int arowIdx = r0 + lm;
    const float* arow;
    if constexpr (GATHER) arow = A + (size_t)gidx[arowIdx] * lda;
    else                  arow = A + (size_t)arowIdx * lda;

    v8f acc[4] = {};

    for (int k0 = 0; k0 < KPAD; k0 += 4) {
        int ka = k0 + 2 * half;       // this lane's first K index
        int kb = ka + 1;
        bool oka, okb;
        if constexpr (KMAX == KPAD) { oka = true; okb = true; }
        else { oka = (ka < KMAX); okb = (kb < KMAX); }

        v2f a;
        a.x = oka ? arow[ka] : 0.0f;
        a.y = okb ? arow[kb] : 0.0f;

        const float* brow0 = Bw + (size_t)ka * ldb + n0 + lm;
        const float* brow1 = Bw + (size_t)kb * ldb + n0 + lm;
#pragma unroll
        for (int t = 0; t < 4; ++t) {
            bool cok = (n0 + t * 16 + lm) < ncolsB;
            v2f b;
            b.x = (oka && cok) ? brow0[t * 16] : 0.0f;
            b.y = (okb && cok) ? brow1[t * 16] : 0.0f;
            acc[t] = wmma_f32(a, b, acc[t]);
        }
    }

#pragma unroll
    for (int t = 0; t < 4; ++t) {
        int col = n0 + t * 16 + lm;
        if (col < ncols) {
            float bb = bias[col];
#pragma unroll
            for (int j = 0; j < 8; ++j) {
                int orow = r0 + j + 8 * half;
                float v = acc[t][j] + bb;
                out[(size_t)orow * ldo + col] = fmaxf(v, 0.0f);
            }
        }
    }
}

// ---------------------------------------------------------------------------
// Codebook squared norms
// ---------------------------------------------------------------------------
__global__ __launch_bounds__(256) void cnorm_kernel(const float* __restrict__ cb,
                                                    float* __restrict__ cnorm)
{
    int k = blockIdx.x * blockDim.x + threadIdx.x;
    if (k >= NCODE) return;
    float s = 0.0f;
    const float* row = cb + (size_t)k * DMODEL;
    for (int d = 0; d < DMODEL; ++d) { float v = row[d]; s += v * v; }
    cnorm[k] = s;
}

// ---------------------------------------------------------------------------
// Deterministic per-batch sum / sumsq partials (fixed-shape LDS tree)
// blockIdx.x = b*chunks + c
// ---------------------------------------------------------------------------
__global__ __launch_bounds__(256) void reduce_stats_kernel(
    const float* __restrict__ src, float* __restrict__ partial,
    int perBatch, int chunks, int perChunk)
{
    int b = blockIdx.x / chunks;
    int c = blockIdx.x - b * chunks;
    const float* p = src + (size_t)b * perBatch + (size_t)c * perChunk;
    float s = 0.0f, q = 0.0f;
    for (int i = threadIdx.x; i < perChunk; i += 256) {
        float v = p[i];
        s += v; q += v * v;
    }
    __shared__ float ss[256];
    __shared__ float sq[256];
    ss[threadIdx.x] = s; sq[threadIdx.x] = q;
    __syncthreads();
    for (int off = 128; off > 0; off >>= 1) {
        if (threadIdx.x < off) {
            ss[threadIdx.x] += ss[threadIdx.x + off];
            sq[threadIdx.x] += sq[threadIdx.x + off];
        }
        __syncthreads();
    }
    if (threadIdx.x == 0) {
        partial[(size_t)blockIdx.x * 2 + 0] = ss[0];
        partial[(size_t)blockIdx.x * 2 + 1] = sq[0];
    }
}

__global__ void finalize_stats_kernel(const float* __restrict__ partial,
                                      float* __restrict__ stats,
                                      int chunks, float invN)
{
    if (threadIdx.x != 0) return;
    int b = blockIdx.x;
    float s = 0.0f, q = 0.0f;
    for (int c = 0; c < chunks; ++c) {          // fixed order -> deterministic
        s += partial[(size_t)(b * chunks + c) * 2 + 0];
        q += partial[(size_t)(b * chunks + c) * 2 + 1];
    }
    float mu  = s * invN;
    float var = fmaxf(q * invN - mu * mu, 0.0f);
    stats[b * 2 + 0] = mu;
    stats[b * 2 + 1] = rsqrtf(var + 1e-5f);
}

// ---------------------------------------------------------------------------
// VQ: one wave per 16-row tile. Normalize into LDS, WMMA distances with the
// codebook as the A matrix (so each lane's 8 accumulator slots are distinct
// codes for one fixed row), in-lane argmin + one shfl_xor(16) merge.
// Also emits the per-tile commit-loss partial.
// ---------------------------------------------------------------------------
#define HN_STRIDE 257
__global__ __launch_bounds__(32) void vq_kernel(
    const float* __restrict__ h2, const float* __restrict__ stats,
    const float* __restrict__ g,  const float* __restrict__ beta,
    const float* __restrict__ cb, const float* __restrict__ cnorm,
    int* __restrict__ indices, float* __restrict__ commit_partial)
{
    __shared__ float hn[16 * HN_STRIDE];
    __shared__ int   idx16[16];

    int lane = threadIdx.x;          // 0..31, single wave
    int r0   = blockIdx.x * 16;

    // normalize 16x256 tile into LDS
    for (int c = 0; c < 128; ++c) {
        int flat = c * 32 + lane;    // 0..4095
        int row  = flat >> 8;
        int col  = flat & 255;
        int r    = r0 + row;
        int b    = r / TLEN;
        int t    = r - b * TLEN;
        float mu   = stats[b * 2 + 0];
        float rstd = stats[b * 2 + 1];
        float v = (h2[(size_t)r * DMODEL + col] - mu) * rstd
                  * g[(size_t)t * DMODEL + col] + beta[(size_t)t * DMODEL + col];
        hn[row * HN_STRIDE + col] = v;
    }
    __syncthreads();

    int half = lane >> 4;
    int lm   = lane & 15;

    float minv = 3.402823466e38f;
    int   mini = 0;

    for (int m = 0; m < 32; ++m) {                 // 32 tiles of 16 codes
        v8f acc = {};
        const float* arow = cb + (size_t)(m * 16 + lm) * DMODEL;
        const float* brow = hn + lm * HN_STRIDE;
        for (int k0 = 0; k0 < DMODEL; k0 += 4) {
            int ka = k0 + 2 * half;
            v2f a, b;
            a.x = arow[ka];     a.y = arow[ka + 1];
            b.x = brow[ka];     b.y = brow[ka + 1];
            acc = wmma_f32(a, b, acc);
        }
#pragma unroll
        for (int j = 0; j < 8; ++j) {
            int code = m * 16 + j + 8 * half;
            float d = cnorm[code] - 2.0f * acc[j];   // |h|^2 constant per row: dropped
            if (d < minv) { minv = d; mini = code; } // ascending code -> first-min ties
        }
    }

    // merge the two code halves (same dist row lives in lanes l and l^16)
    float ov = __shfl_xor(minv, 16, 32);
    int   oi = __shfl_xor(mini, 16, 32);
    if (ov < minv || (ov == minv && oi < mini)) { minv = ov; mini = oi; }

    if (lane < 16) {
        indices[r0 + lane] = mini;
        idx16[lane] = mini;
    }
    __syncthreads();

    // commit loss partial: sum (codebook[idx] - hn)^2 over the tile
    float s = 0.0f;
    for (int c = 0; c < 128; ++c) {
        int flat = c * 32 + lane;
        int row  = flat >> 8;
        int col  = flat & 255;
        float diff = cb[(size_t)idx16[row] * DMODEL + col] - hn[row * HN_STRIDE + col];
        s += diff * diff;
    }
#pragma unroll
    for (int off = 16; off > 0; off >>= 1) s += __shfl_xor(s, off, 32);
    if (lane == 0) commit_partial[blockIdx.x] = s;
}

__global__ void commit_finalize_kernel(const float* __restrict__ partial,
                                       float* __restrict__ out)
{
    if (threadIdx.x == 0 && blockIdx.x == 0) {
        float s = 0.0f;
        for (int i = 0; i < MTILES; ++i) s += partial[i];   // fixed order
        out[0] = s * (1.0f / ((float)BT * (float)DMODEL));  // COMMIT_W = 1
    }
}

// ---------------------------------------------------------------------------
// Final decoder LayerNorm
// ---------------------------------------------------------------------------
__global__ __launch_bounds__(256) void final_ln_kernel(
    const float* __restrict__ y2, const float* __restrict__ stats,
    const float* __restrict__ g,  const float* __restrict__ beta,
    float* __restrict__ out)
{
    int i = blockIdx.x * 256 + threadIdx.x;
    if (i >= YTOTAL) return;
    int b   = i / DEC_PER_BATCH;
    int rem = i - b * DEC_PER_BATCH;
    int t   = rem / DOUT;
    int o   = rem - t * DOUT;
    float mu   = stats[b * 2 + 0];
    float rstd = stats[b * 2 + 1];
    out[i] = (y2[i] - mu) * rstd * g[t * DOUT + o] + beta[t * DOUT + o];
}

// ---------------------------------------------------------------------------
extern "C" void kernel_launch(void* const* d_in, const int* in_sizes, int n_in,
                              void* d_out, int out_size, void* d_ws, size_t ws_size,
                              hipStream_t stream) {
    (void)in_sizes; (void)n_in; (void)out_size; (void)ws_size;
    const float* x        = (const float*)d_in[0];
    const float* enc_w1   = (const float*)d_in[1];
    const float* enc_b1   = (const float*)d_in[2];
    const float* enc_w2   = (const float*)d_in[3];
    const float* enc_b2   = (const float*)d_in[4];
    const float* enc_g    = (const float*)d_in[5];
    const float* enc_beta = (const float*)d_in[6];
    const float* codebook = (const float*)d_in[7];
    const float* dec_w1   = (const float*)d_in[8];
    const float* dec_b1   = (const float*)d_in[9];
    const float* dec_w2   = (const float*)d_in[10];
    const float* dec_b2   = (const float*)d_in[11];
    const float* dec_g    = (const float*)d_in[12];
    const float* dec_beta = (const float*)d_in[13];
    float* out = (float*)d_out;

    float* ws = (float*)d_ws;
    size_t o = 0;
    float* h1     = ws + o; o += (size_t)BT * DMODEL;   // reused as y1
    float* h2     = ws + o; o += (size_t)BT * DMODEL;   // reused as y2
    float* cnorm  = ws + o; o += NCODE;
    float* pA     = ws + o; o += (size_t)NBATCH * ENC_CHUNKS * 2;
    float* stA    = ws + o; o += NBATCH * 2;
    float* pB     = ws + o; o += (size_t)NBATCH * DEC_CHUNKS * 2;
    float* stB    = ws + o; o += NBATCH * 2;
    float* cpart  = ws + o; o += MTILES;
    int*   idxs   = (int*)(ws + o);

    // codebook squared norms
    cnorm_kernel<<<2, 256, 0, stream>>>(codebook, cnorm);

    // encoder GEMM1: h1 = relu(x @ enc_w1 + b1)   (K = 58 padded to 60)
    gemm_relu_kernel<DIN, 60, false><<<MTILES * 4 / 8, 256, 0, stream>>>(
        x, nullptr, enc_w1, enc_b1, h1, DIN, DMODEL, DMODEL, DMODEL, DMODEL, 4);

    // encoder GEMM2: h2 = relu(h1 @ enc_w2 + b2)
    gemm_relu_kernel<DMODEL, DMODEL, false><<<MTILES * 4 / 8, 256, 0, stream>>>(
        h1, nullptr, enc_w2, enc_b2, h2, DMODEL, DMODEL, DMODEL, DMODEL, DMODEL, 4);

    // encoder LN stats (deterministic two-stage)
    reduce_stats_kernel<<<NBATCH * ENC_CHUNKS, 256, 0, stream>>>(
        h2, pA, ENC_PER_BATCH, ENC_CHUNKS, ENC_PER_CHUNK);
    finalize_stats_kernel<<<NBATCH, 32, 0, stream>>>(pA, stA, ENC_CHUNKS,
                                                     1.0f / (float)ENC_PER_BATCH);

    // LN + VQ argmin + commit partials
    vq_kernel<<<MTILES, 32, 0, stream>>>(h2, stA, enc_g, enc_beta, codebook,
                                         cnorm, idxs, cpart);
    commit_finalize_kernel<<<1, 32, 0, stream>>>(cpart, out + YTOTAL);

    // decoder GEMM1: y1 = relu(codebook[idx] @ dec_w1 + b1)  (gathered A)
    gemm_relu_kernel<DMODEL, DMODEL, true><<<MTILES * 4 / 8, 256, 0, stream>>>(
        codebook, idxs, dec_w1, dec_b1, h1, DMODEL, DMODEL, DMODEL, DMODEL, DMODEL, 4);

    // decoder GEMM2: y2 = relu(y1 @ dec_w2 + b2)  (N = 58, one 64-col group)
    gemm_relu_kernel<DMODEL, DMODEL, false><<<MTILES / 8, 256, 0, stream>>>(
        h1, nullptr, dec_w2, dec_b2, h2, DMODEL, DOUT, DOUT, DOUT, DOUT, 1);

    // decoder LN stats + final LN
    reduce_stats_kernel<<<NBATCH * DEC_CHUNKS, 256, 0, stream>>>(
        h2, pB, DEC_PER_BATCH, DEC_CHUNKS, DEC_PER_CHUNK);
    finalize_stats_kernel<<<NBATCH, 32, 0, stream>>>(pB, stB, DEC_CHUNKS,
                                                     1.0f / (float)DEC_PER_BATCH);
    final_ln_kernel<<<(YTOTAL + 255) / 256, 256, 0, stream>>>(h2, stB, dec_g,
                                                              dec_beta, out);
}